// MaskBoundary_77129022702067
// MI455X (gfx1250) — compile-verified
//
#include <hip/hip_runtime.h>

// Boundary mask over int32 label image [8,1,1024,2048].
// Memory-bound stencil. CDNA5 path: async global->LDS B128/B32 staging
// (ASYNCcnt) of a 64x32 tile + halo, LDS b128 window reads, XOR-OR
// boundary reduction, float4 stores. Border blocks take a fully
// predicated slow path matching the reference's edge masking.

#define B_   8
#define H_   1024
#define W_   2048
#define TW   64                    // tile width (pixels)
#define TH   32                    // tile height (pixels)
#define HROWS (TH + 2)             // 34 halo rows
#define LDSW  68                   // LDS row stride in ints (272B, 16B-aligned)
#define NTHREADS 256
#define NB128 (HROWS * 16)         // 544 interior lane-B128 loads per block

typedef int v4i __attribute__((vector_size(4 * sizeof(int))));

typedef __attribute__((address_space(1))) int g_int;   // global (AS1)
typedef __attribute__((address_space(3))) int l_int;   // LDS (AS3)
typedef __attribute__((address_space(1))) v4i g_v4i;   // global vec4 (AS1)
typedef __attribute__((address_space(3))) v4i l_v4i;   // LDS vec4 (AS3)

__device__ __forceinline__ void async_load_b32(const int* gp, int* lp) {
#if __has_builtin(__builtin_amdgcn_global_load_async_to_lds_b32)
    __builtin_amdgcn_global_load_async_to_lds_b32((g_int*)gp, (l_int*)lp, 0, 0);
#else
    unsigned laddr = (unsigned)(unsigned long long)(l_int*)lp;
    asm volatile("global_load_async_to_lds_b32 %0, %1, off"
                 :: "v"(laddr), "v"(gp) : "memory");
#endif
}

__device__ __forceinline__ void async_load_b128(const int* gp, int* lp) {
#if __has_builtin(__builtin_amdgcn_global_load_async_to_lds_b128)
    __builtin_amdgcn_global_load_async_to_lds_b128((g_v4i*)gp, (l_v4i*)lp, 0, 0);
#elif __has_builtin(__builtin_amdgcn_global_load_async_to_lds_b32)
    // INST_OFFSET applies to both global and LDS addresses (ISA 10.x).
    __builtin_amdgcn_global_load_async_to_lds_b32((g_int*)gp, (l_int*)lp, 0, 0);
    __builtin_amdgcn_global_load_async_to_lds_b32((g_int*)gp, (l_int*)lp, 4, 0);
    __builtin_amdgcn_global_load_async_to_lds_b32((g_int*)gp, (l_int*)lp, 8, 0);
    __builtin_amdgcn_global_load_async_to_lds_b32((g_int*)gp, (l_int*)lp, 12, 0);
#else
    unsigned laddr = (unsigned)(unsigned long long)(l_int*)lp;
    asm volatile("global_load_async_to_lds_b128 %0, %1, off"
                 :: "v"(laddr), "v"(gp) : "memory");
#endif
}

__device__ __forceinline__ void wait_async_all() {
#if __has_builtin(__builtin_amdgcn_s_wait_asynccnt)
    __builtin_amdgcn_s_wait_asynccnt(0);
#else
    asm volatile("s_wait_asynccnt 0" ::: "memory");
#endif
}

// 6-wide window of one LDS row: v[i] holds column (j0 - 1 + i).
struct Win { int v[6]; };

__device__ __forceinline__ Win load_win(const int* tile, int row, int j0,
                                        int jl, int jr) {
    Win w;
    const int* rp = tile + row * LDSW;
    int4 m = *(const int4*)(rp + j0);          // ds_load_b128 (16B aligned)
    w.v[0] = rp[jl];
    w.v[1] = m.x; w.v[2] = m.y; w.v[3] = m.z; w.v[4] = m.w;
    w.v[5] = rp[jr];
    return w;
}

// Interior blocks: all 8 neighbors in-bounds -> pure XOR-OR reduction.
__device__ __forceinline__ float4 row_fast(const Win& U, const Win& C, const Win& D) {
    float4 o; float* op = &o.x;
#pragma unroll
    for (int p = 0; p < 4; ++p) {
        const int c = C.v[p + 1];
        const int t = (c ^ C.v[p]) | (c ^ C.v[p + 2])
                    | (c ^ U.v[p]) | (c ^ U.v[p + 1]) | (c ^ U.v[p + 2])
                    | (c ^ D.v[p]) | (c ^ D.v[p + 1]) | (c ^ D.v[p + 2]);
        op[p] = t ? 1.0f : 0.0f;
    }
    return o;
}

// Border blocks: predicate every direction (reference edge masking).
__device__ __forceinline__ float4 row_slow(const Win& U, const Win& C, const Win& D,
                                           int gxb, int gy) {
    const bool vu = gy > 0;
    const bool vd = gy < H_ - 1;
    float4 o; float* op = &o.x;
#pragma unroll
    for (int p = 0; p < 4; ++p) {
        const int  c  = C.v[p + 1];
        const bool vl = (gxb + p) > 0;
        const bool vr = (gxb + p) < W_ - 1;
        const bool bnd =
            (vr &&       c != C.v[p + 2]) ||
            (vl &&       c != C.v[p    ]) ||
            (vd &&       c != D.v[p + 1]) ||
            (vu &&       c != U.v[p + 1]) ||
            (vu && vr && c != U.v[p + 2]) ||
            (vd && vr && c != D.v[p + 2]) ||
            (vu && vl && c != U.v[p    ]) ||
            (vd && vl && c != D.v[p    ]);
        op[p] = bnd ? 1.0f : 0.0f;
    }
    return o;
}

__global__ __launch_bounds__(NTHREADS)
void MaskBoundary_kernel(const int* __restrict__ in, float* __restrict__ out) {
    // Row layout: cols [0..63] interior (16B-aligned), col 64 = right halo,
    // col 65 = left halo. Row stride 272B keeps every interior B128 aligned.
    __shared__ __align__(16) int tile[HROWS * LDSW];

    const int tid = threadIdx.x;
    const int x0  = blockIdx.x * TW;
    const int y0  = blockIdx.y * TH;
    const long plane = (long)blockIdx.z * (H_ * W_);
    const int* gin = in + plane;

    // ---- Stage interior: 34 rows x 16 lane-B128 async loads.
    {
        int i = tid;
#pragma unroll
        for (int pass = 0; pass < 2; ++pass) {        // i, i+256 < 544 always
            const int r = i >> 4;                     // 0..33
            const int k = i & 15;
            int gy = y0 + r - 1;
            gy = gy < 0 ? 0 : (gy > H_ - 1 ? H_ - 1 : gy);
            async_load_b128(gin + (long)gy * W_ + x0 + k * 4,
                            &tile[r * LDSW + k * 4]);
            i += NTHREADS;
        }
        if (tid < NB128 - 2 * NTHREADS) {             // tid < 32
            const int i2 = tid + 2 * NTHREADS;
            const int r = i2 >> 4;
            const int k = i2 & 15;
            int gy = y0 + r - 1;
            gy = gy < 0 ? 0 : (gy > H_ - 1 ? H_ - 1 : gy);
            async_load_b128(gin + (long)gy * W_ + x0 + k * 4,
                            &tile[r * LDSW + k * 4]);
        }
    }
    // ---- Stage halo columns: 34 rows x {left,right} as B32 async loads.
    if (tid < 2 * HROWS) {                            // tid < 68
        const int r    = tid >> 1;
        const int side = tid & 1;                     // 0 = left, 1 = right
        int gy = y0 + r - 1;
        gy = gy < 0 ? 0 : (gy > H_ - 1 ? H_ - 1 : gy);
        int gx = side ? (x0 + TW) : (x0 - 1);
        gx = gx < 0 ? 0 : (gx > W_ - 1 ? W_ - 1 : gx);
        async_load_b32(gin + (long)gy * W_ + gx,
                       &tile[r * LDSW + 65 - side]);  // left->65, right->64
    }
    wait_async_all();     // my wave's async LDS writes have landed
    __syncthreads();      // everyone's have

    // ---- Compute: each thread owns a 4x2 block of pixels.
    const int tx = tid & 15;
    const int ty = tid >> 4;
    const int j0 = tx * 4;
    const int jl = (tx == 0)  ? 65 : j0 - 1;          // wrapped left column
    const int jr = (tx == 15) ? 64 : j0 + 4;          // wrapped right column
    const int r0 = 2 * ty;                            // halo row above output row 0

    const Win w0 = load_win(tile, r0,     j0, jl, jr);
    const Win w1 = load_win(tile, r0 + 1, j0, jl, jr);
    const Win w2 = load_win(tile, r0 + 2, j0, jl, jr);
    const Win w3 = load_win(tile, r0 + 3, j0, jl, jr);

    const int gy0 = y0 + 2 * ty;
    const int gxb = x0 + j0;

    const bool border = (blockIdx.x == 0) | (blockIdx.x == (W_ / TW - 1)) |
                        (blockIdx.y == 0) | (blockIdx.y == (H_ / TH - 1));
    float4 o0, o1;
    if (!border) {                                    // uniform branch
        o0 = row_fast(w0, w1, w2);
        o1 = row_fast(w1, w2, w3);
    } else {
        o0 = row_slow(w0, w1, w2, gxb, gy0);
        o1 = row_slow(w1, w2, w3, gxb, gy0 + 1);
    }

    float* ob = out + plane + (long)gy0 * W_ + gxb;   // 16B aligned
    *(float4*)ob         = o0;
    *(float4*)(ob + W_)  = o1;
}

extern "C" void kernel_launch(void* const* d_in, const int* in_sizes, int n_in,
                              void* d_out, int out_size, void* d_ws, size_t ws_size,
                              hipStream_t stream) {
    (void)in_sizes; (void)n_in; (void)out_size; (void)d_ws; (void)ws_size;
    const int* in  = (const int*)d_in[0];
    float*     out = (float*)d_out;
    dim3 grid(W_ / TW, H_ / TH, B_);   // 32 x 32 x 8
    MaskBoundary_kernel<<<grid, NTHREADS, 0, stream>>>(in, out);
}